// GCN_90477781058259
// MI455X (gfx1250) — compile-verified
//
#include <hip/hip_runtime.h>

typedef float v2f __attribute__((ext_vector_type(2)));
typedef float v8f __attribute__((ext_vector_type(8)));

#define FDIM 128
#define S2 144   // float2 pair-row stride: 2*144 % 64 == 32 -> lane halves hit disjoint banks

// ---------------- gcn_norm kernels ----------------
__global__ __launch_bounds__(256) void k_deg_init(float* __restrict__ deg, int N) {
    int i = blockIdx.x * 256 + threadIdx.x;
    if (i < N) deg[i] = 1.0f;                 // self-loop weight
}

__global__ __launch_bounds__(256) void k_deg_acc(const int* __restrict__ col,
                                                 const float* __restrict__ w,
                                                 float* __restrict__ deg, int E) {
    int e = blockIdx.x * 256 + threadIdx.x;
    if (e < E) atomicAdd(&deg[col[e]], w[e]);
}

__global__ __launch_bounds__(256) void k_dinv(const float* __restrict__ deg,
                                              float* __restrict__ dinv, int N) {
    int i = blockIdx.x * 256 + threadIdx.x;
    if (i < N) {
        float d = deg[i];
        dinv[i] = d > 0.0f ? rsqrtf(d) : 0.0f;
    }
}

__global__ __launch_bounds__(256) void k_norm(const int* __restrict__ row,
                                              const int* __restrict__ col,
                                              const float* __restrict__ w,
                                              const float* __restrict__ dinv,
                                              float* __restrict__ nrm, int E) {
    int e = blockIdx.x * 256 + threadIdx.x;
    if (e < E) nrm[e] = dinv[row[e]] * w[e] * dinv[col[e]];
}

// ---------------- fused GEMM (WMMA f32) + self-loop/bias epilogue ----------------
// H   = act(A) @ W                     (act = ReLU if RELU)
// OUT = dinv^2 * H + bias              (self-loop message + bias, pre-scatter init)
// Each wave: one 16-row strip, all 8 column tiles (16x128 of output).
// W staged in LDS as (k even, k odd) float2 pairs so every B fragment is one ds_load_b64.
template <int RELU>
__global__ __launch_bounds__(256) void k_gemm_wmma(const float* __restrict__ A,
                                                   const float* __restrict__ W,
                                                   const float* __restrict__ bias,
                                                   const float* __restrict__ dinv,
                                                   float* __restrict__ H,
                                                   float* __restrict__ OUT,
                                                   int N) {
    __shared__ float2 sW2[64 * S2];     // sW2[p][n] = (W[2p][n], W[2p+1][n])
    for (int idx = threadIdx.x; idx < 64 * FDIM; idx += 256) {
        int p = idx >> 7, n = idx & 127;
        sW2[p * S2 + n] = make_float2(W[(2 * p) * FDIM + n], W[(2 * p + 1) * FDIM + n]);
    }
    __syncthreads();

    const int wave = threadIdx.x >> 5;
    const int lane = threadIdx.x & 31;
    const int half = lane >> 4;          // selects K pair {0,1} vs {2,3}
    const int mr   = lane & 15;          // M for A-frag, N for B/C/D-frag
    const int m0   = (blockIdx.x * 8 + wave) * 16;
    if (m0 >= N) return;                 // wave-uniform: EXEC stays all-1s for WMMA

    v8f acc[8];
#pragma unroll
    for (int t = 0; t < 8; ++t) acc[t] = (v8f)(0.0f);

    const int   mA  = m0 + mr;
    const bool  inb = mA < N;
    const float msk = inb ? 1.0f : 0.0f;             // zero OOB rows via VALU, not EXEC
    const float* Arow = A + (size_t)(inb ? mA : 0) * FDIM;  // clamped: always valid

    for (int k0 = 0; k0 < FDIM; k0 += 4) {
        const int ka = k0 + 2 * half;                // even -> 8B-aligned b64 load
        v2f a = *(const v2f*)(Arow + ka);
        if (RELU) { a.x = fmaxf(a.x, 0.0f); a.y = fmaxf(a.y, 0.0f); }
        a.x *= msk;
        a.y *= msk;
        const float2* bRow = sW2 + (size_t)((k0 >> 1) + half) * S2 + mr;
#pragma unroll
        for (int t = 0; t < 8; ++t) {
            const float2 bf = bRow[t * 16];          // one ds_load_b64, aligned pair
            v2f b;
            b.x = bf.x;                              // W[ka][n]
            b.y = bf.y;                              // W[ka+1][n]
            acc[t] = __builtin_amdgcn_wmma_f32_16x16x4_f32(
                false, a, false, b, (short)0, acc[t], false, false);
        }
    }

    float bn[8];
#pragma unroll
    for (int t = 0; t < 8; ++t) bn[t] = bias[t * 16 + mr];

#pragma unroll
    for (int v = 0; v < 8; ++v) {
        const int m = m0 + v + 8 * half;   // C/D: row = vreg + 8*half
        if (m < N) {
            const float dv = dinv[m];
            const float s  = dv * dv;      // self-loop norm
            float* hrow = H   + (size_t)m * FDIM;
            float* orow = OUT + (size_t)m * FDIM;
#pragma unroll
            for (int t = 0; t < 8; ++t) {
                const float hv = acc[t][v];
                hrow[t * 16 + mr] = hv;
                orow[t * 16 + mr] = s * hv + bn[t];
            }
        }
    }
}

// ---------------- edge scatter: OUT[col] += norm * H[row] ----------------
// One wave per edge; lane q covers features 4q..4q+3 (float4 gather, 4 atomics).
__global__ __launch_bounds__(256) void k_edge_scatter(const int* __restrict__ row,
                                                      const int* __restrict__ col,
                                                      const float* __restrict__ nrm,
                                                      const float* __restrict__ H,
                                                      float* __restrict__ OUT, int E) {
    const long long idx = (long long)blockIdx.x * 256 + threadIdx.x;
    if (idx >= (long long)E * 32) return;
    const int e = (int)(idx >> 5);
    const int q = (int)(idx & 31);
    const float wn = nrm[e];
    const int r = row[e];
    const int c = col[e];
    const float4 hv = ((const float4*)H)[(size_t)r * 32 + q];
    float* dst = OUT + (size_t)c * FDIM + q * 4;
    atomicAdd(dst + 0, wn * hv.x);
    atomicAdd(dst + 1, wn * hv.y);
    atomicAdd(dst + 2, wn * hv.z);
    atomicAdd(dst + 3, wn * hv.w);
}

extern "C" void kernel_launch(void* const* d_in, const int* in_sizes, int n_in,
                              void* d_out, int out_size, void* d_ws, size_t ws_size,
                              hipStream_t stream) {
    const float* x  = (const float*)d_in[0];
    const int*   ei = (const int*)  d_in[1];
    const float* ew = (const float*)d_in[2];
    const float* W1 = (const float*)d_in[3];
    const float* b1 = (const float*)d_in[4];
    const float* W2 = (const float*)d_in[5];
    const float* b2 = (const float*)d_in[6];
    float* out = (float*)d_out;

    const int N = in_sizes[0] / FDIM;
    const int E = in_sizes[2];
    const int* rowp = ei;        // edge_index[0]
    const int* colp = ei + E;    // edge_index[1]

    // workspace layout (floats)
    float* ws = (float*)d_ws;
    size_t off = 0;
    float* deg  = ws + off; off += (size_t)((N + 3) & ~3);
    float* dinv = ws + off; off += (size_t)((N + 3) & ~3);
    float* nrm  = ws + off; off += (size_t)((E + 3) & ~3);
    float* bufA = ws + off; off += (size_t)N * FDIM;   // h (layer1), then h2
    float* bufB = ws + off;                            // out1 accumulator

    dim3 blk(256);
    const int nb_n = (N + 255) / 256;
    const int nb_e = (E + 255) / 256;
    const int nb_g = (N + 127) / 128;                        // 8 waves x 16 rows
    const long long sth = (long long)E * 32;
    const int nb_s = (int)((sth + 255) / 256);

    // gcn_norm
    k_deg_init<<<nb_n, blk, 0, stream>>>(deg, N);
    k_deg_acc <<<nb_e, blk, 0, stream>>>(colp, ew, deg, E);
    k_dinv    <<<nb_n, blk, 0, stream>>>(deg, dinv, N);
    k_norm    <<<nb_e, blk, 0, stream>>>(rowp, colp, ew, dinv, nrm, E);

    // layer 1: h1 = x@W1 ; out1 = dinv^2*h1 + b1 ; out1 += scatter(norm*h1[row])
    k_gemm_wmma<0><<<nb_g, blk, 0, stream>>>(x, W1, b1, dinv, bufA, bufB, N);
    k_edge_scatter<<<nb_s, blk, 0, stream>>>(rowp, colp, nrm, bufA, bufB, E);

    // layer 2: h2 = relu(out1)@W2 ; out = dinv^2*h2 + b2 ; out += scatter(norm*h2[row])
    k_gemm_wmma<1><<<nb_g, blk, 0, stream>>>(bufB, W2, b2, dinv, bufA, out, N);
    k_edge_scatter<<<nb_s, blk, 0, stream>>>(rowp, colp, nrm, bufA, out, E);
}